// ResGatedGCNConvGraphGymLayer_45286135169320
// MI455X (gfx1250) — compile-verified
//
#include <hip/hip_runtime.h>

#define DIM 128

typedef float v2f __attribute__((ext_vector_type(2)));
typedef float v8f __attribute__((ext_vector_type(8)));

__device__ __forceinline__ v2f ld2(const float* p) { return *(const v2f*)p; }

// ---------------------------------------------------------------------------
// Kernel 1: fused 4-way GEMM  (k = x Wk^T, q = x Wq^T, v = x Wv^T,
//                              out = x Wskip^T + b)
// One block = one 16-row tile of x.  4 waves per block; wave w owns weight w.
// Exact fp32 via V_WMMA_F32_16X16X4_F32 (A 16x4, B 4x16, C/D 16x16 f32).
// ---------------------------------------------------------------------------
__global__ void __launch_bounds__(128) gemm4_kernel(
    const float* __restrict__ x,
    const float* __restrict__ Wk, const float* __restrict__ Wq,
    const float* __restrict__ Wv, const float* __restrict__ Wskip,
    const float* __restrict__ bias,
    float* __restrict__ kout, float* __restrict__ qout,
    float* __restrict__ vout, float* __restrict__ out, int N)
{
    const int lane = threadIdx.x & 31;
    const int wave = threadIdx.x >> 5;          // 0..3 -> Wk,Wq,Wv,Wskip
    const int row_base = blockIdx.x * 16;

    const float* W   = (wave == 0) ? Wk   : (wave == 1) ? Wq   :
                       (wave == 2) ? Wv   : Wskip;
    float*       Dst = (wave == 0) ? kout : (wave == 1) ? qout :
                       (wave == 2) ? vout : out;

    // A-fragment addressing (16x4 f32 layout):
    //   lanes 0-15: rows 0-15, k = {0,1}; lanes 16-31: rows 0-15, k = {2,3}
    const int mrow = lane & 15;
    const int koff = (lane >> 4) << 1;          // 0 or 2

    // Clamp load row for (hypothetical) ragged N; EXEC stays all-ones.
    int arow = row_base + mrow;
    if (arow >= N) arow = N - 1;
    const float* xrow = x + (size_t)arow * DIM + koff;

    v8f acc[8] = {};                            // 8 N-tiles of 16 columns

    #pragma unroll 4
    for (int kk = 0; kk < DIM / 4; ++kk) {      // 32 K-steps of 4
        v2f a = ld2(xrow + kk * 4);
        #pragma unroll
        for (int n = 0; n < 8; ++n) {
            // B-fragment (4x16): B[k][col] = W[col][k]  (out = x @ W^T)
            v2f b = ld2(W + (size_t)(n * 16 + mrow) * DIM + kk * 4 + koff);
            acc[n] = __builtin_amdgcn_wmma_f32_16x16x4_f32(
                false, a, false, b, (short)0, acc[n], false, false);
        }
    }

    // C/D layout: VGPR r -> (M = r + 8*(lane>=16), N = lane&15)
    const int srow0 = row_base + ((lane >> 4) << 3);
    const int scol  = lane & 15;
    #pragma unroll
    for (int n = 0; n < 8; ++n) {
        const int col = n * 16 + scol;
        const float bb = (wave == 3) ? bias[col] : 0.0f;
        #pragma unroll
        for (int r = 0; r < 8; ++r) {
            const int row = srow0 + r;
            if (row < N)
                Dst[(size_t)row * DIM + col] = acc[n][r] + bb;
        }
    }
}

// ---------------------------------------------------------------------------
// Kernel 2: edge scatter.  One wave32 per edge: D=128 = 32 lanes x float4.
// msg = sigmoid(k[dst] + q[src]) * v[src], atomically accumulated into out.
// Working set (k,q,v,out ~ 102 MB) is resident in the 192 MB L2.
// ---------------------------------------------------------------------------
__global__ void __launch_bounds__(256) edge_kernel(
    const float* __restrict__ kf, const float* __restrict__ qf,
    const float* __restrict__ vf, const long long* __restrict__ ei,
    float* __restrict__ out, int E)
{
    const int gwave = (int)((blockIdx.x * blockDim.x + threadIdx.x) >> 5);
    const int lane  = threadIdx.x & 31;
    if (gwave >= E) return;                     // wave-uniform

    const long long s = ei[gwave];              // src (row 0)
    const long long d = ei[(size_t)E + gwave];  // dst (row 1)

    const float4 kv = ((const float4*)(kf + (size_t)d * DIM))[lane];
    const float4 qv = ((const float4*)(qf + (size_t)s * DIM))[lane];
    const float4 vv = ((const float4*)(vf + (size_t)s * DIM))[lane];

    float4 m;
    m.x = vv.x / (1.0f + __expf(-(kv.x + qv.x)));
    m.y = vv.y / (1.0f + __expf(-(kv.y + qv.y)));
    m.z = vv.z / (1.0f + __expf(-(kv.z + qv.z)));
    m.w = vv.w / (1.0f + __expf(-(kv.w + qv.w)));

    float* op = out + (size_t)d * DIM + lane * 4;
    unsafeAtomicAdd(op + 0, m.x);
    unsafeAtomicAdd(op + 1, m.y);
    unsafeAtomicAdd(op + 2, m.z);
    unsafeAtomicAdd(op + 3, m.w);
}

// ---------------------------------------------------------------------------
extern "C" void kernel_launch(void* const* d_in, const int* in_sizes, int n_in,
                              void* d_out, int out_size, void* d_ws, size_t ws_size,
                              hipStream_t stream)
{
    const float*     x     = (const float*)d_in[0];
    const long long* ei    = (const long long*)d_in[1];
    const float*     Wk    = (const float*)d_in[2];
    const float*     Wq    = (const float*)d_in[3];
    const float*     Wv    = (const float*)d_in[4];
    const float*     Wskip = (const float*)d_in[5];
    const float*     bias  = (const float*)d_in[6];
    float*           out   = (float*)d_out;

    const int N = in_sizes[0] / DIM;
    const int E = in_sizes[1] / 2;

    float* ws = (float*)d_ws;
    float* kf = ws;
    float* qf = ws + (size_t)N * DIM;
    float* vf = ws + 2 * (size_t)N * DIM;

    const int tiles = (N + 15) / 16;
    gemm4_kernel<<<tiles, 128, 0, stream>>>(x, Wk, Wq, Wv, Wskip, bias,
                                            kf, qf, vf, out, N);

    const int blocks = (E + 7) / 8;             // 8 waves (edges) per block
    edge_kernel<<<blocks, 256, 0, stream>>>(kf, qf, vf, ei, out, E);
}